// Mu_sigma_26740466385351
// MI455X (gfx1250) — compile-verified
//
#include <hip/hip_runtime.h>

typedef __attribute__((ext_vector_type(2))) float v2f;
typedef __attribute__((ext_vector_type(8))) float v8f;

#define BLOCK 256
#define WAVES 8      // 256 threads / wave32
#define CHUNKS 16    // blocks per batch

// ---------------------------------------------------------------- zero ws
__global__ void __launch_bounds__(BLOCK) musig_zero(float* __restrict__ ws, int n) {
    int i = blockIdx.x * BLOCK + threadIdx.x;
    if (i < n) ws[i] = 0.0f;
}

// ---------------------------------------------------------------- main reduction
// ws layout: ws[b*16 + k*4 + m], m==3 slot holds S0[b,k] (weight * 1.0)
__global__ void __launch_bounds__(BLOCK) musig_partial(
    const float* __restrict__ pred,   // [16,4,XY]
    const float* __restrict__ xin,    // [16,3,XY]
    const int*   __restrict__ heart,  // [16,1,XY]
    float* __restrict__ ws, int XY) {

    __shared__ float red[WAVES][16][33];   // [wave][value][lane] (+1 pad)

    const int tid   = threadIdx.x;
    const int wave  = tid >> 5;
    const int lane  = tid & 31;
    const int b     = blockIdx.x / CHUNKS;
    const int chunk = blockIdx.x % CHUNKS;

    const int chunkSize = (XY + CHUNKS - 1) / CHUNKS;
    const int pBeg = chunk * chunkSize;
    int pEnd = pBeg + chunkSize; if (pEnd > XY) pEnd = XY;

    const float* pr = pred  + (size_t)b * 4 * XY;
    const float* xi = xin   + (size_t)b * 3 * XY;
    const int*   hh = heart + (size_t)b * XY;

    float acc[16];
#pragma unroll
    for (int i = 0; i < 16; ++i) acc[i] = 0.0f;

    // Hot loop: fully coalesced streaming loads + FMA (bandwidth-bound path).
    for (int p = pBeg + tid; p < pEnd; p += BLOCK) {
        const float mk = (hh[p] == 1) ? 1.0f : 0.0f;
        const float w0 = pr[p            ] * mk;
        const float w1 = pr[p +     XY   ] * mk;
        const float w2 = pr[p + 2 * XY   ] * mk;
        const float w3 = pr[p + 3 * XY   ] * mk;
        const float x0 = xi[p            ];
        const float x1 = xi[p +     XY   ];
        const float x2 = xi[p + 2 * XY   ];
        acc[ 0] += w0 * x0; acc[ 1] += w0 * x1; acc[ 2] += w0 * x2; acc[ 3] += w0;
        acc[ 4] += w1 * x0; acc[ 5] += w1 * x1; acc[ 6] += w1 * x2; acc[ 7] += w1;
        acc[ 8] += w2 * x0; acc[ 9] += w2 * x1; acc[10] += w2 * x2; acc[11] += w2;
        acc[12] += w3 * x0; acc[13] += w3 * x1; acc[14] += w3 * x2; acc[15] += w3;
    }

    // Stage per-lane partials: red[wave][value][lane]
#pragma unroll
    for (int v = 0; v < 16; ++v) red[wave][v][lane] = acc[v];
    __syncthreads();

    // Cross-lane reduction as a matmul against ones:
    //   D[i,j] += sum_p A[i,p] * 1   with A[i,p] = red[wave][i][4q+p]
    // 8 chained V_WMMA_F32_16X16X4_F32 reduce all 32 lanes in exact f32.
    // A layout (32b 16x4): lanes 0-15 row=i supply K=0,1; lanes 16-31 supply K=2,3.
    v8f c = {};
    const v2f bones = {1.0f, 1.0f};          // all-ones B is layout-invariant
    const int row  = lane & 15;
    const int half = (lane >> 4) << 1;       // 0 for lanes 0-15, 2 for 16-31
#pragma unroll
    for (int q = 0; q < 8; ++q) {
        v2f a;
        a.x = red[wave][row][4 * q + half];
        a.y = red[wave][row][4 * q + half + 1];
        c = __builtin_amdgcn_wmma_f32_16x16x4_f32(
                false, a, false, bones, (short)0, c, false, false);
    }

    // D layout: VGPR r = row r (lanes 0-15) / row 8+r (lanes 16-31); all cols equal.
    float* wsb = ws + b * 16;
    if (lane == 0) {
#pragma unroll
        for (int r = 0; r < 8; ++r) atomicAdd(&wsb[r], c[r]);
    } else if (lane == 16) {
#pragma unroll
        for (int r = 0; r < 8; ++r) atomicAdd(&wsb[8 + r], c[r]);
    }
}

// ---------------------------------------------------------------- finalize
__global__ void musig_final(const float* __restrict__ ws, float* __restrict__ out) {
    if (threadIdx.x != 0 || blockIdx.x != 0) return;
    const float gt[4][3] = {
        { 1.0907561f, -0.15736851f, 1.70052552f},
        {-0.38956344f, 0.35907112f, 0.00187305f},
        { 0.12236157f, 0.99027221f, 0.42901193f},
        { 1.25641782f, 0.86135793f, 0.53020001f}};
    float s = 0.0f;
    for (int k = 0; k < 4; ++k) {
        for (int m = 0; m < 3; ++m) {
            float mm = 0.0f;
            for (int b = 0; b < 16; ++b) {
                const float S0 = ws[b * 16 + k * 4 + 3];
                const float S1 = ws[b * 16 + k * 4 + m];
                mm += S1 / (S0 + 1e-10f);
            }
            mm *= (1.0f / 16.0f);
            const float d = gt[k][m] - mm;
            s += d * d;
        }
    }
    out[0] = s;
}

// ---------------------------------------------------------------- launcher
extern "C" void kernel_launch(void* const* d_in, const int* in_sizes, int n_in,
                              void* d_out, int out_size, void* d_ws, size_t ws_size,
                              hipStream_t stream) {
    const float* pred  = (const float*)d_in[0];
    const float* xin   = (const float*)d_in[1];
    const int*   heart = (const int*)d_in[2];
    float* ws  = (float*)d_ws;
    float* out = (float*)d_out;

    const int XY = in_sizes[2] / 16;   // heart is [16,1,X,Y]

    musig_zero<<<1, BLOCK, 0, stream>>>(ws, 16 * 16);
    musig_partial<<<16 * CHUNKS, BLOCK, 0, stream>>>(pred, xin, heart, ws, XY);
    musig_final<<<1, 32, 0, stream>>>(ws, out);
}